// FullGapModel_1460288881186
// MI455X (gfx1250) — compile-verified
//
#include <hip/hip_runtime.h>

typedef float v2f __attribute__((ext_vector_type(2)));
typedef float v8f __attribute__((ext_vector_type(8)));

#define D_DIM   400
#define NT      5                    // 16-col tiles per k-pass
#define ACOLS   (NT * 16)            // 80 columns of A staged in LDS
#define KSTEPS  (D_DIM / 4)          // 100 wmma k-steps (K=4 each)
#define NGROUPS (D_DIM / ACOLS)      // 5 column groups
#define WAVES_PER_BLOCK 8
#define BLOCK_THREADS   (WAVES_PER_BLOCK * 32)
#define ENVS_PER_BLOCK  (WAVES_PER_BLOCK * 16)

__global__ void zero_kernel(float* __restrict__ out, int n) {
    int i = blockIdx.x * blockDim.x + threadIdx.x;
    if (i < n) out[i] = 0.0f;
}

// A[e][d] = sum_m weights[col_seg[m]] * sp[m][e] * sp[m][d]   (D x D, D=400)
__global__ void build_A_kernel(const float* __restrict__ sp,
                               const float* __restrict__ w,
                               const int*   __restrict__ col_seg,
                               float* __restrict__ A, int M, int D) {
    __shared__ float s_e[32][16];
    __shared__ float s_d[32][16];
    __shared__ float s_w[32];
    const int tx = threadIdx.x, ty = threadIdx.y;
    const int tid = ty * 16 + tx;
    const int d = blockIdx.x * 16 + tx;
    const int e = blockIdx.y * 16 + ty;
    float acc = 0.0f;
    for (int m0 = 0; m0 < M; m0 += 32) {
        for (int i = tid; i < 32 * 16; i += 256) {
            int mm = i >> 4, cc = i & 15;
            int m = m0 + mm;
            float ve = 0.0f, vd = 0.0f;
            if (m < M) {
                ve = sp[(size_t)m * D + blockIdx.y * 16 + cc];
                vd = sp[(size_t)m * D + blockIdx.x * 16 + cc];
            }
            s_e[mm][cc] = ve;
            s_d[mm][cc] = vd;
        }
        if (tid < 32) {
            int m = m0 + tid;
            s_w[tid] = (m < M) ? w[col_seg[m]] : 0.0f;
        }
        __syncthreads();
#pragma unroll 8
        for (int j = 0; j < 32; ++j)
            acc = fmaf(s_w[j] * s_e[j][ty], s_d[j][tx], acc);
        __syncthreads();
    }
    A[(size_t)e * D + d] = acc;
}

// out[row_seg[i]] += (p_i^T A p_i) / ||p_i||^2  via fp32 WMMA 16x16x4
__global__ void __launch_bounds__(BLOCK_THREADS, 1)
qform_kernel(const float* __restrict__ P, const float* __restrict__ A,
             const int* __restrict__ row_seg, float* __restrict__ out, int N) {
    __shared__ float asl[D_DIM * ACOLS];   // 400*80*4 = 125 KB (2 blocks/WGP in 320 KB LDS)

    const int tid  = threadIdx.x;
    const int lane = tid & 31;
    const int wave = tid >> 5;
    const int half = lane >> 4;    // K-half for A/B operands
    const int n    = lane & 15;    // matrix row (A-op) / col (B-op, C/D)
    const int i0   = blockIdx.x * ENVS_PER_BLOCK + wave * 16;

    int rowA = i0 + n; if (rowA > N - 1) rowA = N - 1;           // clamp; excluded at writeout
    const float2* __restrict__ p2 =
        reinterpret_cast<const float2*>(P + (size_t)rowA * D_DIM);

    float dotacc[8];
#pragma unroll
    for (int r = 0; r < 8; ++r) dotacc[r] = 0.0f;
    float n2acc = 0.0f;   // accumulated NGROUPS times; rescaled at the end

    for (int g = 0; g < NGROUPS; ++g) {
        __syncthreads();
        // cooperatively stage A[:, g*80 .. g*80+79] into LDS, row-major 80-wide
        for (int idx = tid; idx < D_DIM * ACOLS; idx += BLOCK_THREADS) {
            int kk = idx / ACOLS, cc = idx % ACOLS;
            asl[idx] = A[(size_t)kk * D_DIM + g * ACOLS + cc];
        }
        __syncthreads();

        v8f cacc[NT];
#pragma unroll
        for (int t = 0; t < NT; ++t)
            cacc[t] = (v8f){0.f, 0.f, 0.f, 0.f, 0.f, 0.f, 0.f, 0.f};

#pragma unroll 2
        for (int ks = 0; ks < KSTEPS; ++ks) {
            // A-operand (16x4 f32): lanes 0-15 -> K = 4ks+{0,1}; lanes 16-31 -> K = 4ks+{2,3}
            float2 av = p2[2 * ks + half];
            v2f a; a.x = av.x; a.y = av.y;
            n2acc = fmaf(av.x, av.x, fmaf(av.y, av.y, n2acc));  // unconditional; /NGROUPS later
            const int kk = 4 * ks + 2 * half;
            const float* b0 = &asl[kk * ACOLS + n];
            const float* b1 = &asl[(kk + 1) * ACOLS + n];
#pragma unroll
            for (int t = 0; t < NT; ++t) {
                v2f b; b.x = b0[t * 16]; b.y = b1[t * 16];   // B-operand (4x16 f32)
                cacc[t] = __builtin_amdgcn_wmma_f32_16x16x4_f32(
                    false, a, false, b, (short)0, cacc[t], false, false);
            }
        }

        // row-dot: C/D layout -> VGPR r holds env (r + 8*half), column n
#pragma unroll
        for (int t = 0; t < NT; ++t) {
            const int dbase = g * ACOLS + t * 16 + n;
#pragma unroll
            for (int r = 0; r < 8; ++r) {
                int row = i0 + r + 8 * half; if (row > N - 1) row = N - 1;
                float pv = P[(size_t)row * D_DIM + dbase];   // cache-hot re-read
                dotacc[r] = fmaf(pv, cacc[t][r], dotacc[r]);
            }
        }
    }

    // butterfly-reduce each env's dot across the 16 lanes of its half
#pragma unroll
    for (int r = 0; r < 8; ++r) {
        float v = dotacc[r];
        v += __shfl_xor(v, 1);
        v += __shfl_xor(v, 2);
        v += __shfl_xor(v, 4);
        v += __shfl_xor(v, 8);
        dotacc[r] = v;
    }
    // ||p||^2: lane L and L^16 each hold half the K range for env (L&15);
    // each was accumulated NGROUPS times over identical data -> rescale.
    float n2full = (n2acc + __shfl_xor(n2acc, 16)) * (1.0f / (float)NGROUPS);
    // writing lane (half*16 + r), r<8, handles env r + 8*half; fetch its norm
    float n2e = __shfl(n2full, (lane & 15) + 8 * half);

    float q = 0.0f;
#pragma unroll
    for (int r = 0; r < 8; ++r)
        if ((lane & 15) == r) q = dotacc[r];

    const int env = (lane & 15) + 8 * half;
    const int row = i0 + env;
    if ((lane & 15) < 8 && row < N)
        atomicAdd(&out[row_seg[row]], q / n2e);
}

extern "C" void kernel_launch(void* const* d_in, const int* in_sizes, int n_in,
                              void* d_out, int out_size, void* d_ws, size_t ws_size,
                              hipStream_t stream) {
    const float* P       = (const float*)d_in[0];   // [N, D]
    const float* SP      = (const float*)d_in[1];   // [M, D]
    const float* W       = (const float*)d_in[2];   // [1, T]
    const int*   row_seg = (const int*)  d_in[4];   // [N]
    const int*   col_seg = (const int*)  d_in[5];   // [M]
    float* out = (float*)d_out;                     // [S]

    const int N = in_sizes[4];
    const int M = in_sizes[5];
    const int D = in_sizes[0] / N;                  // == 400 == D_DIM
    const int S = out_size;

    float* Amat = (float*)d_ws;                     // D*D floats = 640 KB scratch

    zero_kernel<<<(S + 255) / 256, 256, 0, stream>>>(out, S);
    dim3 gA(D / 16, D / 16);
    build_A_kernel<<<gA, dim3(16, 16), 0, stream>>>(SP, W, col_seg, Amat, M, D);
    const int blocks = (N + ENVS_PER_BLOCK - 1) / ENVS_PER_BLOCK;
    qform_kernel<<<blocks, BLOCK_THREADS, 0, stream>>>(P, Amat, row_seg, out, N);
}